// HANLayer_24240795419353
// MI455X (gfx1250) — compile-verified
//
#include <hip/hip_runtime.h>
#include <hip/hip_bf16.h>
#include <math.h>

// Problem constants (match reference)
constexpr int NN  = 30000;   // nodes
constexpr int EE  = 300000;  // edges per metapath
constexpr int INF_ = 128;    // input features
constexpr int HH  = 4;       // heads
constexpr int DD  = 64;      // dim per head
constexpr int HD  = 256;     // H*D
constexpr int SEM = 128;     // semantic hidden
constexpr float NEG_ATTN = 0.2f;
constexpr float NEG_ACT  = 0.01f;

typedef float v2f __attribute__((ext_vector_type(2)));
typedef float v8f __attribute__((ext_vector_type(8)));

// -------------------- workspace layout (floats) --------------------
constexpr size_t FEAT_OFF = 0;                          // 4*N*HD
constexpr size_t FEAT_SZ  = 4ull * NN * HD;
constexpr size_t GAT_OFF  = FEAT_OFF + FEAT_SZ;         // 4*N*HD (accum, then activated in place)
constexpr size_t EL_OFF   = GAT_OFF + FEAT_SZ;          // 4*N*H
constexpr size_t NH_SZ    = 4ull * NN * HH;
constexpr size_t ER_OFF   = EL_OFF + NH_SZ;
constexpr size_t M_OFF    = ER_OFF + NH_SZ;
constexpr size_t DEN_OFF  = M_OFF + NH_SZ;
constexpr size_t S_OFF    = DEN_OFF + NH_SZ;            // 4 scalars
constexpr size_t BETA_OFF = S_OFF + 4;                  // 4 scalars

// -------------------- helpers --------------------
__device__ __forceinline__ void atomicMaxFloat(float* addr, float val) {
  unsigned int* ua = (unsigned int*)addr;
  unsigned int old = __float_as_uint(*addr);
  while (true) {
    float cur = __uint_as_float(old);
    if (cur >= val) break;
    unsigned int assumed = old;
    old = atomicCAS(ua, assumed, __float_as_uint(val));
    if (old == assumed) break;
  }
}

__global__ void fill_neginf_kernel(float* p, int n) {
  int i = blockIdx.x * blockDim.x + threadIdx.x;
  if (i < n) p[i] = -INFINITY;
}

// -------------------- 1) feat = hs[m] @ W[m]  (WMMA f32 16x16x4) --------------------
// grid: (N/16, 4), block: 256 (8 waves). Each wave computes two 16x16 tiles.
__global__ void gat_gemm_kernel(const float* __restrict__ hs,
                                const float* __restrict__ W,
                                float* __restrict__ feat) {
  const int m       = blockIdx.y;
  const int rowBase = blockIdx.x * 16;
  const int tid  = threadIdx.x;
  const int wave = tid >> 5;
  const int lane = tid & 31;

  __shared__ float As[16][INF_ + 4];   // pad 4 -> bank stride 4, conflict-free

  const float* A = hs + ((size_t)m * NN + rowBase) * INF_;
  for (int i = tid; i < 16 * (INF_ / 4); i += 256) {
    int r  = i / (INF_ / 4);
    int c4 = i % (INF_ / 4);
    float4 v = ((const float4*)(A + (size_t)r * INF_))[c4];
    ((float4*)&As[r][0])[c4] = v;
  }
  __syncthreads();

  const int row   = lane & 15;   // A row / B,C,D column within tile
  const int khalf = lane >> 4;   // which K pair this lane holds
  const float* Bm = W + (size_t)m * INF_ * HD;
  float* out = feat + (size_t)m * NN * HD;

  for (int t = 0; t < 2; ++t) {
    const int colBase = (wave + t * 8) * 16;   // 16 tiles cover all 256 cols
    v8f acc = {};
#pragma unroll
    for (int k0 = 0; k0 < INF_; k0 += 4) {
      v2f a, b;
      a.x = As[row][k0 + khalf * 2 + 0];
      a.y = As[row][k0 + khalf * 2 + 1];
      const float* bp = Bm + (size_t)(k0 + khalf * 2) * HD + colBase + row;
      b.x = bp[0];
      b.y = bp[HD];
      acc = __builtin_amdgcn_wmma_f32_16x16x4_f32(false, a, false, b,
                                                  (short)0, acc, false, false);
    }
#pragma unroll
    for (int r = 0; r < 8; ++r) {
      int orow = rowBase + r + khalf * 8;
      out[(size_t)orow * HD + colBase + row] = acc[r];
    }
  }
}

// -------------------- 2) el/er = sum_d feat*attn --------------------
// grid: (N, 4), block 128 (one wave per head; 64 dims -> 2 per lane)
__global__ void attn_dot_kernel(const float* __restrict__ feat,
                                const float* __restrict__ attn_l,
                                const float* __restrict__ attn_r,
                                float* __restrict__ el, float* __restrict__ er) {
  const int n = blockIdx.x, m = blockIdx.y;
  const int t = threadIdx.x;
  const int h = t >> 5, lane = t & 31;
  const float* f  = feat + ((size_t)m * NN + n) * HD + h * DD;
  const float* al = attn_l + ((size_t)m * HH + h) * DD;
  const float* ar = attn_r + ((size_t)m * HH + h) * DD;
  float f0 = f[lane], f1 = f[lane + 32];
  float pl = f0 * al[lane] + f1 * al[lane + 32];
  float pr = f0 * ar[lane] + f1 * ar[lane + 32];
#pragma unroll
  for (int off = 16; off > 0; off >>= 1) {
    pl += __shfl_down(pl, off);
    pr += __shfl_down(pr, off);
  }
  if (lane == 0) {
    el[((size_t)m * NN + n) * HH + h] = pl;
    er[((size_t)m * NN + n) * HH + h] = pr;
  }
}

// -------------------- 3) segment max of leaky(el[src]+er[dst]) --------------------
__global__ void edge_max_kernel(const int* __restrict__ src, const int* __restrict__ dst,
                                const float* __restrict__ el, const float* __restrict__ er,
                                float* __restrict__ mbuf) {
  size_t idx = (size_t)blockIdx.x * blockDim.x + threadIdx.x;  // over 4*E
  if (idx >= 4ull * EE) return;
  int m = (int)(idx / EE), e = (int)(idx % EE);
  int s = src[m * EE + e], d = dst[m * EE + e];
#pragma unroll
  for (int h = 0; h < HH; ++h) {
    float v = el[((size_t)m * NN + s) * HH + h] + er[((size_t)m * NN + d) * HH + h];
    v = (v > 0.f) ? v : NEG_ATTN * v;
    atomicMaxFloat(&mbuf[((size_t)m * NN + d) * HH + h], v);
  }
}

__global__ void m_fixup_kernel(float* __restrict__ mbuf, int n) {
  int i = blockIdx.x * blockDim.x + threadIdx.x;
  if (i < n) { float v = mbuf[i]; mbuf[i] = isfinite(v) ? v : 0.f; }
}

// -------------------- 4) den = segment_sum exp(e - m[dst]) --------------------
__global__ void edge_den_kernel(const int* __restrict__ src, const int* __restrict__ dst,
                                const float* __restrict__ el, const float* __restrict__ er,
                                const float* __restrict__ mbuf, float* __restrict__ den) {
  size_t idx = (size_t)blockIdx.x * blockDim.x + threadIdx.x;
  if (idx >= 4ull * EE) return;
  int m = (int)(idx / EE), e = (int)(idx % EE);
  int s = src[m * EE + e], d = dst[m * EE + e];
#pragma unroll
  for (int h = 0; h < HH; ++h) {
    float v = el[((size_t)m * NN + s) * HH + h] + er[((size_t)m * NN + d) * HH + h];
    v = (v > 0.f) ? v : NEG_ATTN * v;
    float ex = expf(v - mbuf[((size_t)m * NN + d) * HH + h]);
    atomicAdd(&den[((size_t)m * NN + d) * HH + h], ex);
  }
}

// -------------------- 5) out[dst] += feat[src] * alpha --------------------
// grid: (E, 4), block 256 (one thread per channel)
__global__ void edge_agg_kernel(const int* __restrict__ src, const int* __restrict__ dst,
                                const float* __restrict__ el, const float* __restrict__ er,
                                const float* __restrict__ mbuf, const float* __restrict__ den,
                                const float* __restrict__ feat, float* __restrict__ outacc) {
  const int e = blockIdx.x, m = blockIdx.y;
  const int c = threadIdx.x, h = c >> 6;
  const int s = src[m * EE + e], d = dst[m * EE + e];
  float v = el[((size_t)m * NN + s) * HH + h] + er[((size_t)m * NN + d) * HH + h];
  v = (v > 0.f) ? v : NEG_ATTN * v;
  float ex = expf(v - mbuf[((size_t)m * NN + d) * HH + h]);
  float alpha = ex / fmaxf(den[((size_t)m * NN + d) * HH + h], 1e-9f);
  float msg = feat[((size_t)m * NN + s) * HD + c] * alpha;
  atomicAdd(&outacc[((size_t)m * NN + d) * HD + c], msg);
}

// -------------------- 6) activation: gat = leaky_0.01(acc + bias) (in place) ------
__global__ void act_bias_kernel(float* __restrict__ gat, const float* __restrict__ bias) {
  size_t idx = (size_t)blockIdx.x * blockDim.x + threadIdx.x;  // over 4*N*HD
  if (idx >= FEAT_SZ) return;
  int m = (int)(idx / ((size_t)NN * HD));
  int c = (int)(idx % HD);
  float g = gat[idx] + bias[m * HD + c];
  gat[idx] = (g > 0.f) ? g : NEG_ACT * g;
}

// -------------------- 7) semantic scores s[j] = sum_n tanh(gat[j]@Wp1+bp1)@Wp2 ----
// grid: (N/16, 4), block 256 (8 waves: one 16-col tile each, K=256), WMMA f32
__global__ void sem_score_kernel(const float* __restrict__ gat,
                                 const float* __restrict__ Wp1,
                                 const float* __restrict__ bp1,
                                 const float* __restrict__ Wp2,
                                 float* __restrict__ s4) {
  const int j       = blockIdx.y;
  const int rowBase = blockIdx.x * 16;
  const int tid  = threadIdx.x;
  const int wave = tid >> 5;
  const int lane = tid & 31;

  __shared__ float As[16][HD + 4];
  __shared__ float blocksum;
  if (tid == 0) blocksum = 0.f;

  const float* A = gat + ((size_t)j * NN + rowBase) * HD;
  for (int i = tid; i < 16 * (HD / 4); i += 256) {
    int r  = i / (HD / 4);
    int c4 = i % (HD / 4);
    float4 v = ((const float4*)(A + (size_t)r * HD))[c4];
    ((float4*)&As[r][0])[c4] = v;
  }
  __syncthreads();

  const int row   = lane & 15;
  const int khalf = lane >> 4;
  const int colBase = wave * 16;   // 8 waves cover SEM=128 cols

  v8f acc = {};
#pragma unroll
  for (int k0 = 0; k0 < HD; k0 += 4) {
    v2f a, b;
    a.x = As[row][k0 + khalf * 2 + 0];
    a.y = As[row][k0 + khalf * 2 + 1];
    const float* bp = Wp1 + (size_t)(k0 + khalf * 2) * SEM + colBase + row;
    b.x = bp[0];
    b.y = bp[SEM];
    acc = __builtin_amdgcn_wmma_f32_16x16x4_f32(false, a, false, b,
                                                (short)0, acc, false, false);
  }

  const int c = colBase + row;
  const float w2 = Wp2[c];
  const float b1 = bp1[c];
  float local = 0.f;
#pragma unroll
  for (int r = 0; r < 8; ++r) local += tanhf(acc[r] + b1) * w2;
#pragma unroll
  for (int off = 16; off > 0; off >>= 1) local += __shfl_down(local, off);
  if (lane == 0) atomicAdd(&blocksum, local);
  __syncthreads();
  if (tid == 0) atomicAdd(&s4[j], blocksum);
}

// -------------------- 8) betas from scores --------------------
__global__ void beta_kernel(const float* __restrict__ s4, float* __restrict__ betas) {
  if (threadIdx.x != 0 || blockIdx.x != 0) return;
  float w0 = s4[0] / NN, w1 = s4[1] / NN, w2 = s4[2] / NN, w3 = s4[3] / NN;
  // lnc over (outs[1], outs[2])
  float mL = fmaxf(w1, w2);
  float e1 = expf(w1 - mL), e2 = expf(w2 - mL), sL = e1 + e2;
  betas[0] = e1 / sL; betas[1] = e2 / sL;
  // dis over (outs[0], outs[3])
  float mD = fmaxf(w0, w3);
  float e0 = expf(w0 - mD), e3 = expf(w3 - mD), sD = e0 + e3;
  betas[2] = e0 / sD; betas[3] = e3 / sD;
}

// -------------------- 9) final combine --------------------
__global__ void combine_kernel(const float* __restrict__ gat,
                               const float* __restrict__ betas,
                               float* __restrict__ out) {
  size_t idx = (size_t)blockIdx.x * blockDim.x + threadIdx.x;  // over N*HD
  if (idx >= (size_t)NN * HD) return;
  float bL1 = betas[0], bL2 = betas[1], bD0 = betas[2], bD3 = betas[3];
  const size_t stride = (size_t)NN * HD;
  out[idx]          = bL1 * gat[1 * stride + idx] + bL2 * gat[2 * stride + idx];
  out[stride + idx] = bD0 * gat[0 * stride + idx] + bD3 * gat[3 * stride + idx];
}

// -------------------- launch --------------------
extern "C" void kernel_launch(void* const* d_in, const int* in_sizes, int n_in,
                              void* d_out, int out_size, void* d_ws, size_t ws_size,
                              hipStream_t stream) {
  (void)in_sizes; (void)n_in; (void)out_size; (void)ws_size;
  const float* hs     = (const float*)d_in[0];
  const int*   src    = (const int*)d_in[1];
  const int*   dst    = (const int*)d_in[2];
  const float* W      = (const float*)d_in[3];
  const float* attn_l = (const float*)d_in[4];
  const float* attn_r = (const float*)d_in[5];
  const float* bias   = (const float*)d_in[6];
  const float* Wp1    = (const float*)d_in[7];
  const float* bp1    = (const float*)d_in[8];
  const float* Wp2    = (const float*)d_in[9];
  float* out = (float*)d_out;
  float* ws  = (float*)d_ws;

  float* feat  = ws + FEAT_OFF;
  float* gat   = ws + GAT_OFF;
  float* el    = ws + EL_OFF;
  float* er    = ws + ER_OFF;
  float* mbuf  = ws + M_OFF;
  float* den   = ws + DEN_OFF;
  float* s4    = ws + S_OFF;
  float* betas = ws + BETA_OFF;

  // init accumulators (graph-capture-safe async memsets)
  hipMemsetAsync(gat, 0, FEAT_SZ * sizeof(float), stream);
  hipMemsetAsync(den, 0, NH_SZ * sizeof(float), stream);
  hipMemsetAsync(s4,  0, 4 * sizeof(float), stream);
  fill_neginf_kernel<<<(int)((NH_SZ + 255) / 256), 256, 0, stream>>>(mbuf, (int)NH_SZ);

  // GAT pipeline
  gat_gemm_kernel<<<dim3(NN / 16, 4), 256, 0, stream>>>(hs, W, feat);
  attn_dot_kernel<<<dim3(NN, 4), 128, 0, stream>>>(feat, attn_l, attn_r, el, er);

  const int edgeBlocks = (int)((4ull * EE + 255) / 256);
  edge_max_kernel<<<edgeBlocks, 256, 0, stream>>>(src, dst, el, er, mbuf);
  m_fixup_kernel<<<(int)((NH_SZ + 255) / 256), 256, 0, stream>>>(mbuf, (int)NH_SZ);
  edge_den_kernel<<<edgeBlocks, 256, 0, stream>>>(src, dst, el, er, mbuf, den);
  edge_agg_kernel<<<dim3(EE, 4), 256, 0, stream>>>(src, dst, el, er, mbuf, den, feat, gat);
  act_bias_kernel<<<(int)((FEAT_SZ + 255) / 256), 256, 0, stream>>>(gat, bias);

  // semantic attention
  sem_score_kernel<<<dim3(NN / 16, 4), 256, 0, stream>>>(gat, Wp1, bp1, Wp2, s4);
  beta_kernel<<<1, 32, 0, stream>>>(s4, betas);
  combine_kernel<<<(int)(((size_t)NN * HD + 255) / 256), 256, 0, stream>>>(gat, betas, out);
}